// TangentPatchNCELoss_55035710931136
// MI455X (gfx1250) — compile-verified
//
#include <hip/hip_runtime.h>
#include <cmath>

typedef __attribute__((ext_vector_type(2))) float v2f;
typedef __attribute__((ext_vector_type(8))) float v8f;

namespace {

constexpr int CH = 3, H = 64, W = 64;
constexpr int D = CH * H * W;            // 12288 flattened feature dim
constexpr int KT = 13;                   // tangent columns (+1 diff col, padded to 16)
constexpr int CHUNK = 256;               // d positions staged per block iteration
constexpr int STRIDE = CHUNK + 2;        // LDS row stride (floats): keeps b64 readers off one bank
constexpr float EPSF = 1e-6f;
constexpr float INV_NCE_T = 1.0f / 0.07f;

// Uniform (branch-free) computation of the 13 tangent columns at position
// d = c*4096 + h*64 + w. Columns are unnormalized and stripped of constant
// positive per-column scales (1/(2*eps), 0.1) — column normalization makes
// those irrelevant. Returns the center value so the caller can form the diff.
__device__ __forceinline__ float compute_cols(const float* __restrict__ f,
                                              float x, float y, int c, int h, int w,
                                              float mean, float* __restrict__ v) {
  const int cb = c << 12;
  const int rb = cb + (h << 6);
  const int p  = (h << 6) + w;
  // roll(+1) picks index-1 (wrap), roll(-1) picks index+1 (wrap)
  const float fxm = f[rb + ((w + 63) & 63)];
  const float fxp = f[rb + ((w + 1) & 63)];
  const float fym = f[cb + (((h + 63) & 63) << 6) + w];
  const float fyp = f[cb + (((h + 1) & 63) << 6) + w];
  const float f0 = f[p], f1 = f[4096 + p], f2 = f[8192 + p];
  const float ctr = (c == 0) ? f0 : ((c == 1) ? f1 : f2);
  const float lum = 0.299f * f0 + 0.587f * f1 + 0.114f * f2;
  const float a = fxm - fxp;   // ~ dx
  const float b = fym - fyp;   // ~ dy
  v[0]  = a;
  v[1]  = b;
  v[2]  = -y * a + x * b;      // rotation
  v[3]  = x * a + y * b;       // scale
  v[4]  = x * a;               // scale_x
  v[5]  = y * b;               // scale_y
  v[6]  = y * a;               // shear_x
  v[7]  = x * b;               // shear_y
  v[8]  = x * x * a;           // persp_x
  v[9]  = y * y * b;           // persp_y
  v[10] = ctr;                 // brightness
  v[11] = ctr - mean;          // contrast
  v[12] = ctr - lum;           // saturation
  return ctr;
}

__global__ __launch_bounds__(256)
void tangent_nce_kernel(const float* __restrict__ featq,
                        const float* __restrict__ featk,
                        float* __restrict__ out) {
  __shared__ float sTq[16 * STRIDE];  // staged tangent tile for q: [col][d_local]
  __shared__ float sTk[16 * STRIDE];  // staged tangent tile for k
  __shared__ float sG[2 * 256];       // 16x16 Gram (q, then k)
  __shared__ float sMean[6];          // channel sums: q0..2, k0..2
  __shared__ float sProj[2];

  const int tid  = threadIdx.x;
  const int lane = tid & 31;
  const int wv   = tid >> 5;          // 8 wave32s per block
  const int n    = blockIdx.x;
  const float* fq = featq + (size_t)n * D;
  const float* fk = featk + (size_t)n * D;

  // -------- one-time zeroing: Gram accumulators, means, padding rows 14/15 ----
  sG[tid]       = 0.0f;
  sG[tid + 256] = 0.0f;
  if (tid < 6) sMean[tid] = 0.0f;
  for (int i = tid; i < 2 * STRIDE; i += 256) {
    sTq[14 * STRIDE + i] = 0.0f;
    sTk[14 * STRIDE + i] = 0.0f;
  }
  __syncthreads();

  // -------- pass 1: channel sums (also warms WGP$/L2) --------
  for (int c = 0; c < 3; ++c) {
    const float* bq = fq + (c << 12);
    const float* bk = fk + (c << 12);
    float s0 = 0.0f, s1 = 0.0f;
    for (int i = tid; i < H * W; i += 256) { s0 += bq[i]; s1 += bk[i]; }
    atomicAdd(&sMean[c], s0);       // ds_add_f32
    atomicAdd(&sMean[3 + c], s1);
  }
  __syncthreads();

  // -------- pass 2: Gram of [T | diff] via V_WMMA_F32_16X16X4_F32 ------------
  // A (16x4 fp32): lanes 0-15 hold M=0..15 at K=0(.x),K=1(.y); lanes 16-31 at K=2,3.
  // G = M^T M is symmetric, so the same per-lane float2 serves as both A and B.
  v8f accq = {};
  v8f acck = {};
  const int col   = lane & 15;          // Gram row/col index this lane feeds
  const int dhalf = (lane >> 4) << 1;   // 0 for lanes 0-15, 2 for lanes 16-31

  for (int dbase = 0; dbase < D; dbase += CHUNK) {
    // ---- produce: each thread fills one d position of both tiles (uniform) ----
    const int d = dbase + tid;
    const int c = d >> 12;
    const int h = (d >> 6) & 63;
    const int w = d & 63;
    const float x = -1.0f + (2.0f / 63.0f) * (float)w;
    const float y = -1.0f + (2.0f / 63.0f) * (float)h;

    if (dbase + CHUNK < D) {  // prefetch next chunk (global_prefetch_b8)
      __builtin_prefetch(&fq[d + CHUNK], 0, 1);
      __builtin_prefetch(&fk[d + CHUNK], 0, 1);
    }

    float vq[14], vk[14];
    const float ctrq = compute_cols(fq, x, y, c, h, w,
                                    sMean[c] * (1.0f / 4096.0f), vq);
    const float ctrk = compute_cols(fk, x, y, c, h, w,
                                    sMean[3 + c] * (1.0f / 4096.0f), vk);
    const float diff = ctrq - ctrk;     // exact, unscaled diff column
    vq[13] = diff;
    vk[13] = diff;
#pragma unroll
    for (int i = 0; i < 14; ++i) {
      sTq[i * STRIDE + tid] = vq[i];
      sTk[i * STRIDE + tid] = vk[i];
    }
    __syncthreads();

    // ---- consume: each wave WMMA-accumulates its 32-d slice of the tile ----
#pragma unroll
    for (int s = 0; s < 8; ++s) {
      const int da = (wv << 5) + (s << 2) + dhalf;
      const v2f aq = *reinterpret_cast<const v2f*>(&sTq[col * STRIDE + da]);
      const v2f ak = *reinterpret_cast<const v2f*>(&sTk[col * STRIDE + da]);
      accq = __builtin_amdgcn_wmma_f32_16x16x4_f32(
          false, aq, false, aq, (short)0, accq, false, false);
      acck = __builtin_amdgcn_wmma_f32_16x16x4_f32(
          false, ak, false, ak, (short)0, acck, false, false);
    }
    __syncthreads();  // tile free for next produce phase
  }

  // C/D layout: VGPR r -> M = r (+8 for lanes 16-31), N = lane&15.
  const int m0 = (lane >> 4) << 3;
#pragma unroll
  for (int r = 0; r < 8; ++r) {
    atomicAdd(&sG[(m0 + r) * 16 + col],       accq[r]);
    atomicAdd(&sG[256 + (m0 + r) * 16 + col], acck[r]);
  }
  __syncthreads();

  // -------- finalize: normalize Gram, 13x13 Cholesky, projection distance ----
  if (tid < 2) {  // tid 0 -> q basis, tid 1 -> k basis
    const float* G = sG + tid * 256;
    float nrm[KT];
#pragma unroll
    for (int k = 0; k < KT; ++k) nrm[k] = fmaxf(sqrtf(G[k * 16 + k]), 1e-12f);

    float A[KT][KT], v[KT];
    for (int k = 0; k < KT; ++k) {
      for (int l = 0; l < KT; ++l) A[k][l] = G[k * 16 + l] / (nrm[k] * nrm[l]);
      A[k][k] += EPSF;
      v[k] = G[k * 16 + 13] / nrm[k];
    }
    // In-place Cholesky A = L L^T (lower)
    for (int k = 0; k < KT; ++k) {
      float diag = A[k][k];
      for (int p = 0; p < k; ++p) diag -= A[k][p] * A[k][p];
      diag = sqrtf(fmaxf(diag, 1e-30f));
      A[k][k] = diag;
      const float inv = 1.0f / diag;
      for (int r = k + 1; r < KT; ++r) {
        float s = A[r][k];
        for (int p = 0; p < k; ++p) s -= A[r][p] * A[k][p];
        A[r][k] = s * inv;
      }
    }
    // v^T A^-1 v = || L^-1 v ||^2  via forward solve
    float z[KT];
    float quad = 0.0f;
    for (int k = 0; k < KT; ++k) {
      float s = v[k];
      for (int p = 0; p < k; ++p) s -= A[k][p] * z[p];
      z[k] = s / A[k][k];
      quad += z[k] * z[k];
    }
    const float dd = G[13 * 16 + 13];  // ||diff||^2
    sProj[tid] = dd - quad;
  }
  __syncthreads();

  if (tid == 0) {
    const float dmin = fminf(sProj[0], sProj[1]);
    const float l = -dmin * INV_NCE_T;
    // logsumexp([l, l]) - l  (analytically log 2; computed per the reference recipe)
    const float m   = l;
    const float lse = logf(expf(l - m) + expf(l - m)) + m;
    out[n] = lse - l;
  }
}

}  // namespace

extern "C" void kernel_launch(void* const* d_in, const int* in_sizes, int n_in,
                              void* d_out, int out_size, void* d_ws, size_t ws_size,
                              hipStream_t stream) {
  const float* fq = (const float*)d_in[0];
  const float* fk = (const float*)d_in[1];
  float* out = (float*)d_out;
  const int n = in_sizes[0] / D;  // 256 samples
  tangent_nce_kernel<<<dim3(n), dim3(256), 0, stream>>>(fq, fk, out);
  (void)n_in; (void)out_size; (void)d_ws; (void)ws_size;
}